// DumbMCMC_53790170415132
// MI455X (gfx1250) — compile-verified
//
#include <hip/hip_runtime.h>

// Problem constants (from reference setup_inputs)
#define CHAIN 40960
#define NW    1024
#define NSUB  10
#define NSAMP (CHAIN / NSUB)          // 4096 output samples
#define SORT_THREADS 256              // 8 waves of 32 (wave32)
#define SCAN_THREADS 1024
#define PER_THREAD   (CHAIN / SCAN_THREADS)  // 40 elements per scan thread

#ifndef __has_builtin
#define __has_builtin(x) 0
#endif

// gfx1250 async Global->LDS DMA (ASYNCcnt path). Fallback = plain b128 load.
#define TRY_ASYNC 1
#if TRY_ASYNC && defined(__AMDGCN__) && \
    __has_builtin(__builtin_amdgcn_global_load_async_to_lds_b128)
#define HAVE_ASYNC 1
#else
#define HAVE_ASYNC 0
#endif

#if HAVE_ASYNC
// Builtin signature (from hipcc diagnostic): takes
//   (int __vector_size__(16) AS(1)*, int __vector_size__(16) AS(3)*, imm, imm)
typedef int v4i_vs __attribute__((vector_size(16)));
typedef __attribute__((address_space(1))) v4i_vs as1_v4i;
typedef __attribute__((address_space(3))) v4i_vs as3_v4i;
__device__ __forceinline__ void wait_asynccnt0() {
#if __has_builtin(__builtin_amdgcn_s_wait_asynccnt)
  __builtin_amdgcn_s_wait_asynccnt(0);
#else
  asm volatile("s_wait_asynccnt 0x0" ::: "memory");
#endif
}
#endif

typedef __attribute__((ext_vector_type(2))) float v2f;
typedef __attribute__((ext_vector_type(8))) float v8f;

// ---------------------------------------------------------------------------
// Register compare-exchange for the bitonic network.
// ---------------------------------------------------------------------------
__device__ __forceinline__ void ce_swap(unsigned long long& x,
                                        unsigned long long& y, bool up) {
  if ((x > y) == up) { unsigned long long t = x; x = y; y = t; }
}

// ---------------------------------------------------------------------------
// Load one noise row (1024 f32) and stably argsort it in LDS.
// Key = (float_bits << 32) | index : noise in [0,1) is non-negative so the
// IEEE bit pattern is order-preserving; index in the low bits makes the
// comparison a total order identical to JAX's stable argsort.
// Stages with j<=2 act inside one thread's 4 contiguous elements and are
// executed in registers (k=2 and k=4 entirely; for k>=8 the trailing j=2,j=1
// pair becomes a single LDS round-trip), cutting LDS passes 55 -> 36.
// ---------------------------------------------------------------------------
__device__ __forceinline__ void load_sort_row(unsigned long long* s_key,
                                              float* s_raw,
                                              const float* __restrict__ noise_row,
                                              unsigned tid) {
  const unsigned b = tid * 4u;
#if HAVE_ASYNC
  // Each lane DMAs its own 16B straight into LDS: global_load_async_to_lds_b128
  __builtin_amdgcn_global_load_async_to_lds_b128(
      (as1_v4i*)(noise_row + b), (as3_v4i*)(s_raw + b), 0, 0);
  wait_asynccnt0();           // lane reads only its own LDS slot -> no barrier
  __syncthreads();            // + compiler ordering fence
  const float4 v = reinterpret_cast<const float4*>(s_raw)[tid];
#else
  (void)s_raw;
  const float4 v = reinterpret_cast<const float4*>(noise_row)[tid];  // b128
#endif
  unsigned long long e0 = ((unsigned long long)__float_as_uint(v.x) << 32) | (b + 0);
  unsigned long long e1 = ((unsigned long long)__float_as_uint(v.y) << 32) | (b + 1);
  unsigned long long e2 = ((unsigned long long)__float_as_uint(v.z) << 32) | (b + 2);
  unsigned long long e3 = ((unsigned long long)__float_as_uint(v.w) << 32) | (b + 3);

  // k=2 stage (j=1): directions alternate inside the quad
  ce_swap(e0, e1, true);
  ce_swap(e2, e3, false);
  // k=4 stage (j=2 then j=1): direction uniform per quad
  {
    const bool up = ((b & 4u) == 0u);
    ce_swap(e0, e2, up); ce_swap(e1, e3, up);
    ce_swap(e0, e1, up); ce_swap(e2, e3, up);
  }
  s_key[b + 0] = e0; s_key[b + 1] = e1; s_key[b + 2] = e2; s_key[b + 3] = e3;

  for (unsigned k = 8; k <= NW; k <<= 1) {
    // cross-thread stages: j = k/2 ... 4 through LDS
    for (unsigned j = k >> 1; j >= 4; j >>= 1) {
      __syncthreads();
      for (unsigned i = tid; i < NW; i += SORT_THREADS) {
        const unsigned ixj = i ^ j;
        if (ixj > i) {
          const unsigned long long a = s_key[i];
          const unsigned long long c = s_key[ixj];
          const bool up = ((i & k) == 0);
          if ((a > c) == up) { s_key[i] = c; s_key[ixj] = a; }  // keys unique
        }
      }
    }
    // fused j=2 and j=1 stages in registers (one LDS round-trip of own quad)
    __syncthreads();
    e0 = s_key[b + 0]; e1 = s_key[b + 1]; e2 = s_key[b + 2]; e3 = s_key[b + 3];
    {
      const bool up = ((b & k) == 0u);
      ce_swap(e0, e2, up); ce_swap(e1, e3, up);
      ce_swap(e0, e1, up); ce_swap(e2, e3, up);
    }
    s_key[b + 0] = e0; s_key[b + 1] = e1; s_key[b + 2] = e2; s_key[b + 3] = e3;
  }
  __syncthreads();
}

// ---------------------------------------------------------------------------
// Exact f32 wave-level sum of 32 per-lane partials on the matrix pipe:
// A(16x4): lanes 0-15 vgpr0 = K0, lanes 16-31 vgpr0 = K2 (K1,K3 zeroed);
// B(4x16) = ones  ->  D[m][n] = p[m] + p[m+16].
// Sum the 8 accumulator VGPRs, then shfl_xor(16) merges the two M-halves.
// ---------------------------------------------------------------------------
__device__ __forceinline__ float wave_sum_wmma(float p) {
  v2f a; a[0] = p;    a[1] = 0.0f;
  v2f b; b[0] = 1.0f; b[1] = 1.0f;
  v8f c = {};
  c = __builtin_amdgcn_wmma_f32_16x16x4_f32(false, a, false, b,
                                            (short)0, c, false, false);
  float r = c[0] + c[1] + c[2] + c[3] + c[4] + c[5] + c[6] + c[7];
  r += __shfl_xor(r, 16, 32);
  return r;  // every lane holds the full 32-lane sum
}

// ---------------------------------------------------------------------------
// Kernel 1: per-chain score w[c] = sum bigram[p_i, p_{i+1}] + start/end
// ---------------------------------------------------------------------------
__global__ void __launch_bounds__(SORT_THREADS)
score_kernel(const float* __restrict__ bigram, const float* __restrict__ start,
             const float* __restrict__ endv,   const float* __restrict__ noise,
             float* __restrict__ w) {
  __shared__ unsigned long long s_key[NW];
  __shared__ float s_raw[NW];
  __shared__ float s_sum;
  const unsigned tid = threadIdx.x;
  const unsigned c = blockIdx.x;
  if (tid == 0) s_sum = 0.0f;
  load_sort_row(s_key, s_raw, noise + (size_t)c * NW, tid);

  float partial = 0.0f;
  for (unsigned t = tid; t < NW - 1; t += SORT_THREADS) {
    const unsigned p0 = (unsigned)s_key[t];       // low 32 bits = index
    const unsigned p1 = (unsigned)s_key[t + 1];
    partial += bigram[(size_t)p0 * NW + p1];      // random gather, L2-resident
  }
  if (tid == 0)
    partial += start[(unsigned)s_key[0]] + endv[(unsigned)s_key[NW - 1]];

  const float wsum = wave_sum_wmma(partial);      // v_wmma_f32_16x16x4_f32
  if ((tid & 31u) == 0) atomicAdd(&s_sum, wsum);  // 8 waves -> ds_add_f32
  __syncthreads();
  if (tid == 0) w[c] = s_sum;
}

// ---------------------------------------------------------------------------
// Kernel 2: MH accept chain as an associative function scan.
// f(x) = (x < T) ? C : x  with C < T is closed under composition:
//   (a then b): if b.T > a.T -> b ; else (a.T, a.C < b.T ? (b.C,b.j) : (a.C,a.j))
// ---------------------------------------------------------------------------
struct Fn { float t; float c; int j; };

__device__ __forceinline__ Fn fn_combine(const Fn& a, const Fn& b) {
  Fn r;
  if (b.t > a.t) { r = b; }
  else {
    r.t = a.t;
    if (a.c < b.t) { r.c = b.c; r.j = b.j; }
    else           { r.c = a.c; r.j = a.j; }
  }
  return r;
}

__global__ void __launch_bounds__(SCAN_THREADS)
scan_kernel(const float* __restrict__ w, const float* __restrict__ u,
            int* __restrict__ sel) {
  __shared__ Fn s_fn[SCAN_THREADS];
  const int tid = threadIdx.x;
  const int i0 = tid * PER_THREAD;

  // Serial composition of this thread's 40 step-functions (i >= 1 only)
  Fn agg; agg.t = -3.4e38f; agg.c = -3.4e38f; agg.j = -1;   // identity
  for (int t = 0; t < PER_THREAD; ++t) {
    const int i = i0 + t;
    if (i >= 1) {
      const float wi = w[i];
      const float lu = logf(fmaxf(u[i], 1e-12f));
      Fn f; f.t = wi - lu; f.c = wi; f.j = i;
      agg = fn_combine(agg, f);
    }
  }
  s_fn[tid] = agg;
  __syncthreads();

  // Hillis-Steele inclusive scan over 1024 aggregates
  for (int off = 1; off < SCAN_THREADS; off <<= 1) {
    const bool has = (tid >= off);
    Fn p; if (has) p = s_fn[tid - off];
    Fn v = s_fn[tid];
    __syncthreads();
    if (has) s_fn[tid] = fn_combine(p, v);
    __syncthreads();
  }

  // Exclusive prefix -> starting state, then replay 40 steps, emitting every
  // 10th accepted-state index (js at i % 10 == 9).
  Fn pre;
  if (tid == 0) { pre.t = -3.4e38f; pre.c = 0.0f; pre.j = -1; }
  else          { pre = s_fn[tid - 1]; }
  float x = w[0]; int j = 0;
  if (x < pre.t) { x = pre.c; j = pre.j; }
  for (int t = 0; t < PER_THREAD; ++t) {
    const int i = i0 + t;
    if (i >= 1) {
      const float wi = w[i];
      const float lu = logf(fmaxf(u[i], 1e-12f));
      if (x < wi - lu) { x = wi; j = i; }
    }
    if ((i % NSUB) == (NSUB - 1)) sel[i / NSUB] = j;
  }
}

// ---------------------------------------------------------------------------
// Kernel 3: re-sort only the 4096 selected chains and emit permutations.
// ---------------------------------------------------------------------------
__global__ void __launch_bounds__(SORT_THREADS)
emit_kernel(const float* __restrict__ noise, const int* __restrict__ sel,
            int* __restrict__ out) {
  __shared__ unsigned long long s_key[NW];
  __shared__ float s_raw[NW];
  const unsigned tid = threadIdx.x;
  const unsigned s = blockIdx.x;
  const int chain = sel[s];
  load_sort_row(s_key, s_raw, noise + (size_t)chain * NW, tid);
  for (unsigned t = tid; t < NW; t += SORT_THREADS)
    out[(size_t)s * NW + t] = (int)(unsigned)s_key[t];
}

// ---------------------------------------------------------------------------
extern "C" void kernel_launch(void* const* d_in, const int* in_sizes, int n_in,
                              void* d_out, int out_size, void* d_ws, size_t ws_size,
                              hipStream_t stream) {
  (void)in_sizes; (void)n_in; (void)out_size; (void)ws_size;
  const float* bigram = (const float*)d_in[1];
  const float* start  = (const float*)d_in[2];
  const float* endv   = (const float*)d_in[3];
  const float* noise  = (const float*)d_in[4];
  const float* u      = (const float*)d_in[5];

  float* w   = (float*)d_ws;                                        // 40960 f32
  int*   sel = (int*)((char*)d_ws + (size_t)CHAIN * sizeof(float)); // 4096 i32

  score_kernel<<<CHAIN, SORT_THREADS, 0, stream>>>(bigram, start, endv, noise, w);
  scan_kernel <<<1,     SCAN_THREADS, 0, stream>>>(w, u, sel);
  emit_kernel <<<NSAMP, SORT_THREADS, 0, stream>>>(noise, sel, (int*)d_out);
}